// CopyGenerator_10660108829535
// MI455X (gfx1250) — compile-verified
//
#include <hip/hip_runtime.h>
#include <math.h>

typedef float v2f __attribute__((ext_vector_type(2)));
typedef float v8f __attribute__((ext_vector_type(8)));

constexpr int TT      = 64;
constexpr int BB      = 32;
constexpr int SS      = 50;
constexpr int MROWS   = TT * BB;    // 2048
constexpr int KDIM    = 512;
constexpr int VOC     = 50000;
constexpr int COPYTOK = 4;
constexpr int PADTOK  = 0;
constexpr float EPSV  = 1e-10f;

// Block tile: 64 M rows x 256 N cols. 8 waves, each wave: 4 M-tiles x 2 N-tiles.
constexpr int BM   = 64;
constexpr int NBLK = 256;
constexpr int GX   = (VOC + NBLK - 1) / NBLK;   // 196
constexpr int GY   = MROWS / BM;                // 32
constexpr int AFRAG = (KDIM / 4) * 64;          // 8192 floats per 16-row M-tile

// ---------------------------------------------------------------------------
// K1: logits = hidden @ W^T + b via V_WMMA_F32_16X16X4_F32.
// Writes exp(logit) to out (COPY column -> 0, its logit captured),
// and a deterministic per-(row, block) partial sum of exp to Zpart.
// 64 M rows per block => W fragments fetched from L2 are reused 4x more
// than a 16-row block (32 FLOP per L2 byte in the inner loop).
// ---------------------------------------------------------------------------
__global__ __launch_bounds__(256) void gemm_exp_kernel(
    const float* __restrict__ hidden, const float* __restrict__ Wm,
    const float* __restrict__ bias, float* __restrict__ out,
    float* __restrict__ Zpart, float* __restrict__ copyLogit)
{
    __shared__ float Alds[4 * AFRAG];   // 128 KB, WMMA A-fragment order, 4 M-tiles
    __shared__ float Wsum[8][BM];

    const int tid  = threadIdx.x;
    const int wave = tid >> 5;
    const int lane = tid & 31;
    const int half = lane >> 4;
    const int l16  = lane & 15;
    const int m0   = blockIdx.y * BM;

    // Stage A (64 rows x 512 K) into LDS, pre-swizzled into WMMA A-fragment
    // order per 16-row M-tile:
    //   f = mt*8192 + kk*64 + lane*2 + j  <->  M = mt*16 + (lane&15),
    //                                          K = kk*4 + (lane>>4)*2 + j
    for (int f = tid; f < 4 * AFRAG; f += 256) {
        int mt = f >> 13;
        int g  = f & (AFRAG - 1);
        int kk = g >> 6;
        int l  = (g >> 1) & 31;
        int j  = g & 1;
        Alds[f] = hidden[(size_t)(m0 + mt * 16 + (l & 15)) * KDIM
                         + kk * 4 + ((l >> 4) << 1) + j];
    }
    __syncthreads();

    const int n0 = blockIdx.x * NBLK + wave * 32;

    // VOC % 16 == 0, so each 16-wide tile is fully valid or fully invalid,
    // and validity is uniform across the wave (EXEC stays all-ones for WMMA).
    bool tvalid[2];
    const float* wp[2];
#pragma unroll
    for (int nt = 0; nt < 2; ++nt) {
        int nbase  = n0 + nt * 16;
        tvalid[nt] = (nbase < VOC);
        int ncl    = tvalid[nt] ? (nbase + l16) : 0;   // clamp dead-tile address
        wp[nt]     = Wm + (size_t)ncl * KDIM + (half << 1);
    }

    v8f acc[4][2];
#pragma unroll
    for (int mt = 0; mt < 4; ++mt)
#pragma unroll
        for (int nt = 0; nt < 2; ++nt)
#pragma unroll
            for (int i = 0; i < 8; ++i) acc[mt][nt][i] = 0.0f;

    const float* abase = &Alds[lane * 2];
#pragma unroll 2
    for (int kk = 0; kk < KDIM / 4; ++kk) {
        v2f a[4];
#pragma unroll
        for (int mt = 0; mt < 4; ++mt)
            a[mt] = *(const v2f*)(abase + mt * AFRAG + kk * 64);
        v2f bf[2];
#pragma unroll
        for (int nt = 0; nt < 2; ++nt)
            bf[nt] = *(const v2f*)(wp[nt] + kk * 4);
#pragma unroll
        for (int mt = 0; mt < 4; ++mt)
#pragma unroll
            for (int nt = 0; nt < 2; ++nt)
                acc[mt][nt] = __builtin_amdgcn_wmma_f32_16x16x4_f32(
                    false, a[mt], false, bf[nt], (short)0, acc[mt][nt],
                    false, false);
    }

    // Epilogue: bias, exp, COPY masking, store, per-row partial sums.
    float rsum[4][8];
#pragma unroll
    for (int mt = 0; mt < 4; ++mt)
#pragma unroll
        for (int i = 0; i < 8; ++i) rsum[mt][i] = 0.0f;

#pragma unroll
    for (int nt = 0; nt < 2; ++nt) {
        if (!tvalid[nt]) continue;
        int n    = n0 + nt * 16 + l16;
        float bv = bias[n];
#pragma unroll
        for (int mt = 0; mt < 4; ++mt) {
#pragma unroll
            for (int i = 0; i < 8; ++i) {
                float logit = acc[mt][nt][i] + bv;   // VGPR i -> M = i + half*8
                int r = m0 + mt * 16 + (half << 3) + i;
                float e = expf(logit);
                if (n == COPYTOK) { copyLogit[r] = logit; e = 0.0f; }
                out[(size_t)r * VOC + n] = e;
                rsum[mt][i] += e;
            }
        }
    }

    // Deterministic width-16 shuffle reduction over N, then sequential cross-wave sum.
#pragma unroll
    for (int mt = 0; mt < 4; ++mt) {
#pragma unroll
        for (int i = 0; i < 8; ++i) {
            float v = rsum[mt][i];
            v += __shfl_down(v, 8, 16);
            v += __shfl_down(v, 4, 16);
            v += __shfl_down(v, 2, 16);
            v += __shfl_down(v, 1, 16);
            if (l16 == 0) Wsum[wave][mt * 16 + (half << 3) + i] = v;
        }
    }
    __syncthreads();
    if (tid < BM) {
        float z = 0.0f;
#pragma unroll
        for (int w = 0; w < 8; ++w) z += Wsum[w][tid];
        Zpart[(size_t)(m0 + tid) * GX + blockIdx.x] = z;
    }
}

// ---------------------------------------------------------------------------
// K2: Z[r] = sum of the GX per-block partials (deterministic sequential sum).
// ---------------------------------------------------------------------------
__global__ void zreduce_kernel(const float* __restrict__ Zpart, float* __restrict__ Z)
{
    int r = blockIdx.x * blockDim.x + threadIdx.x;
    if (r >= MROWS) return;
    float z = 0.0f;
    for (int i = 0; i < GX; ++i) z += Zpart[(size_t)r * GX + i];
    Z[r] = z;
}

// ---------------------------------------------------------------------------
// K3: scatter-add in exp domain; one thread per row -> deterministic,
// duplicate target indices handled by the sequential per-row loop.
// exp-domain value v such that v * (1-copy)/Z == attn * copy.
// ---------------------------------------------------------------------------
__global__ void scatter_kernel(
    const float* __restrict__ attn, const int* __restrict__ src,
    const int* __restrict__ alignment, const float* __restrict__ Z,
    const float* __restrict__ copyLogit, float* __restrict__ out)
{
    int r = blockIdx.x * blockDim.x + threadIdx.x;
    if (r >= MROWS) return;
    int b = r % BB;
    float copy  = 1.0f / (1.0f + expf(-copyLogit[r]));
    float scale = copy * Z[r] / (1.0f - copy);
    float* row = out + (size_t)r * VOC;
    for (int s = 0; s < SS; ++s) {
        int v = alignment[src[b * SS + s]];
        row[v] += attn[r * SS + s] * scale;
    }
}

// ---------------------------------------------------------------------------
// K4: per-row exact sum (matches reference out.sum), norm excludes the PAD
// slot (it is replaced by eps), then log(out/norm + eps).
// ---------------------------------------------------------------------------
__global__ __launch_bounds__(256) void finalize_kernel(
    const float* __restrict__ Z, const float* __restrict__ copyLogit,
    float* __restrict__ out)
{
    int r = blockIdx.x;
    float* row = out + (size_t)r * VOC;

    __shared__ float red[256];
    __shared__ float sh_norm, sh_s;

    float sum = 0.0f;
    for (int v = threadIdx.x; v < VOC; v += 256) sum += row[v];
    red[threadIdx.x] = sum;
    __syncthreads();
    for (int stride = 128; stride > 0; stride >>= 1) {
        if (threadIdx.x < stride) red[threadIdx.x] += red[threadIdx.x + stride];
        __syncthreads();
    }
    if (threadIdx.x == 0) {
        float copy = 1.0f / (1.0f + expf(-copyLogit[r]));
        float s    = (1.0f - copy) / Z[r];            // exp-domain -> prob-domain
        float norm = (red[0] - row[PADTOK]) * s + EPSV;
        sh_norm = norm;
        sh_s    = s;
    }
    __syncthreads();
    float norm = sh_norm, s = sh_s;
    for (int v = threadIdx.x; v < VOC; v += 256) {
        float val = (v == PADTOK) ? EPSV : row[v] * s;
        row[v] = logf(val / norm + EPSV);
    }
}

// ---------------------------------------------------------------------------
extern "C" void kernel_launch(void* const* d_in, const int* in_sizes, int n_in,
                              void* d_out, int out_size, void* d_ws, size_t ws_size,
                              hipStream_t stream)
{
    const float* hidden    = (const float*)d_in[0];   // (2048, 512)
    const float* attn      = (const float*)d_in[1];   // (2048, 50)
    const float* Wm        = (const float*)d_in[2];   // (50000, 512)
    const float* bias      = (const float*)d_in[3];   // (50000,)
    const int*   src       = (const int*)d_in[4];     // (32, 50)
    const int*   alignment = (const int*)d_in[5];     // (50000,)
    float* out = (float*)d_out;                       // (2048, 50000)

    float* Zpart = (float*)d_ws;                      // 2048 * 196 floats
    float* Zrow  = Zpart + (size_t)MROWS * GX;        // 2048 floats
    float* copyL = Zrow + MROWS;                      // 2048 floats

    gemm_exp_kernel<<<dim3(GX, GY), 256, 0, stream>>>(hidden, Wm, bias, out, Zpart, copyL);
    zreduce_kernel<<<(MROWS + 255) / 256, 256, 0, stream>>>(Zpart, Zrow);
    scatter_kernel<<<(MROWS + 255) / 256, 256, 0, stream>>>(attn, src, alignment, Zrow, copyL, out);
    finalize_kernel<<<MROWS, 256, 0, stream>>>(Zrow, copyL, out);
}